// NewGCN_78795470013089
// MI455X (gfx1250) — compile-verified
//
#include <hip/hip_runtime.h>

typedef __attribute__((ext_vector_type(2))) float v2f;
typedef __attribute__((ext_vector_type(8))) float v8f;

#define FDIM 128

// ---------------- degree / norm precompute ----------------

__global__ void k_deg_init(float* deg, int n) {
  int i = blockIdx.x * blockDim.x + threadIdx.x;
  if (i < n) deg[i] = 1.0f;  // self-loop weight
}

__global__ void k_deg_acc(const long long* __restrict__ dst,
                          const float* __restrict__ w, float* deg, int e) {
  int i = blockIdx.x * blockDim.x + threadIdx.x;
  if (i < e) unsafeAtomicAdd(&deg[(int)dst[i]], w[i]);
}

__global__ void k_dinv(float* deg, int n) {
  int i = blockIdx.x * blockDim.x + threadIdx.x;
  if (i < n) {
    float d = deg[i];
    deg[i] = (d > 0.0f) ? (1.0f / sqrtf(d)) : 0.0f;  // in place: deg -> dinv
  }
}

__global__ void k_norm(const long long* __restrict__ src,
                       const long long* __restrict__ dst,
                       const float* __restrict__ w,
                       const float* __restrict__ dinv,
                       float* __restrict__ norm, int e) {
  int i = blockIdx.x * blockDim.x + threadIdx.x;
  if (i < e) norm[i] = dinv[(int)src[i]] * w[i] * dinv[(int)dst[i]];
}

// ---------------- aggregation ----------------

// a[n][f] = dinv[n]^2 * g[n][f] + bias[f]   (self-loop contribution + bias)
__global__ void k_agg_init(const float* __restrict__ g,
                           const float* __restrict__ dinv,
                           const float* __restrict__ bias,
                           float* __restrict__ a, int total) {
  int i = blockIdx.x * blockDim.x + threadIdx.x;
  if (i < total) {
    int node = i >> 7;
    int f = i & (FDIM - 1);
    float dv = dinv[node];
    a[i] = dv * dv * g[i] + bias[f];
  }
}

// one wave (32 lanes) per edge: gather full 128-float row as float4, scatter-add
__global__ __launch_bounds__(256)
void k_scatter(const float* __restrict__ h, const float* __restrict__ norm,
               const long long* __restrict__ src, const long long* __restrict__ dst,
               float* __restrict__ out, int e) {
  int gw = (int)((blockIdx.x * blockDim.x + threadIdx.x) >> 5);
  int lane = threadIdx.x & 31;
  if (gw >= e) return;
  float c = norm[gw];
  long long s = src[gw];
  long long d = dst[gw];
  const float4* hv = (const float4*)(h + s * (long long)FDIM);
  float4 v = hv[lane];
  float* o = out + d * (long long)FDIM + lane * 4;
  unsafeAtomicAdd(o + 0, c * v.x);
  unsafeAtomicAdd(o + 1, c * v.y);
  unsafeAtomicAdd(o + 2, c * v.z);
  unsafeAtomicAdd(o + 3, c * v.w);
}

// ---------------- WMMA f32 GEMM:  C[M,128] = act(A[M,128]) * W[128,128] (+bias) ----------------
// block = 128 threads (4 waves); block computes 16 rows x 128 cols.
// wave w computes cols [w*32, w*32+32) as two 16x16 WMMA accumulators.
// A frag (16x4 f32): VGPR0 = K0 (lanes0-15) / K2 (lanes16-31), VGPR1 = K1/K3  (ISA 7.12.2)
// B frag (4x16 f32): same K split across lane halves, N = lane&15.
template <int RELU_IN>
__global__ __launch_bounds__(128)
void k_gemm_wmma(const float* __restrict__ A, const float* __restrict__ Wm,
                 const float* __restrict__ bias, float* __restrict__ C, int M) {
  const int lane = threadIdx.x & 31;
  const int wave = threadIdx.x >> 5;
  const int hi   = lane >> 4;      // 0: lanes 0-15, 1: lanes 16-31
  const int l15  = lane & 15;
  const int m0   = blockIdx.x * 16;
  const int n0   = wave * 32;

  int arow = m0 + l15;
  if (arow > M - 1) arow = M - 1;  // clamp loads; stores guarded below
  const float* Arow = A + (size_t)arow * FDIM;

  v8f c0 = {};
  v8f c1 = {};

#pragma unroll 8
  for (int k0 = 0; k0 < FDIM; k0 += 4) {
    const int kk = k0 + hi * 2;
    v2f a;
    a.x = Arow[kk];
    a.y = Arow[kk + 1];
    if (RELU_IN) {
      a.x = fmaxf(a.x, 0.0f);
      a.y = fmaxf(a.y, 0.0f);
    }
    const float* W0 = Wm + (size_t)kk * FDIM;
    v2f b0, b1;
    b0.x = W0[n0 + l15];
    b0.y = W0[FDIM + n0 + l15];
    b1.x = W0[n0 + 16 + l15];
    b1.y = W0[FDIM + n0 + 16 + l15];
    c0 = __builtin_amdgcn_wmma_f32_16x16x4_f32(false, a, false, b0, (short)0, c0,
                                               false, false);
    c1 = __builtin_amdgcn_wmma_f32_16x16x4_f32(false, a, false, b1, (short)0, c1,
                                               false, false);
  }

  // C/D layout: VGPR r -> row m0 + r + hi*8, col n0 + (lane&15)
  const int col0 = n0 + l15;
  float badd0 = bias ? bias[col0] : 0.0f;
  float badd1 = bias ? bias[col0 + 16] : 0.0f;
#pragma unroll
  for (int r = 0; r < 8; ++r) {
    int row = m0 + r + hi * 8;
    if (row < M) {
      C[(size_t)row * FDIM + col0]      = c0[r] + badd0;
      C[(size_t)row * FDIM + col0 + 16] = c1[r] + badd1;
    }
  }
}

// ---------------- host launch ----------------

extern "C" void kernel_launch(void* const* d_in, const int* in_sizes, int n_in,
                              void* d_out, int out_size, void* d_ws, size_t ws_size,
                              hipStream_t stream) {
  const float* x   = (const float*)d_in[0];
  const float* ew  = (const float*)d_in[1];
  const float* W1  = (const float*)d_in[2];
  const float* b1  = (const float*)d_in[3];
  const float* W2  = (const float*)d_in[4];
  const float* b2  = (const float*)d_in[5];
  const float* Wfc = (const float*)d_in[6];
  const float* bfc = (const float*)d_in[7];
  const long long* ei = (const long long*)d_in[8];

  const int N = in_sizes[0] / FDIM;  // 50000
  const int E = in_sizes[1];         // 800000
  const long long* src = ei;
  const long long* dst = ei + E;

  // workspace layout (all 16B-aligned for these sizes)
  float* dinv = (float*)d_ws;               // N
  float* norm = dinv + N;                   // E
  float* bufG = norm + E;                   // N*128  (GEMM output h)
  float* bufA = bufG + (size_t)N * FDIM;    // N*128  (aggregation buffer)
  float* outF = (float*)d_out;

  const int T = 256;
  const int gN  = (N + T - 1) / T;
  const int gE  = (E + T - 1) / T;
  const int gNF = (N * FDIM + T - 1) / T;
  const int gEdgeWave = (E * 32 + T - 1) / T;  // one wave per edge
  const int gGemm = (N + 15) / 16;

  // 1) symmetric normalization
  k_deg_init<<<gN, T, 0, stream>>>(dinv, N);
  k_deg_acc<<<gE, T, 0, stream>>>(dst, ew, dinv, E);
  k_dinv<<<gN, T, 0, stream>>>(dinv, N);
  k_norm<<<gE, T, 0, stream>>>(src, dst, ew, dinv, norm, E);

  // 2) layer 1: g1 = x @ W1 ; a1 = segsum(norm * g1[src]) + selfloop + b1
  k_gemm_wmma<0><<<gGemm, 128, 0, stream>>>(x, W1, nullptr, bufG, N);
  k_agg_init<<<gNF, T, 0, stream>>>(bufG, dinv, b1, bufA, N * FDIM);
  k_scatter<<<gEdgeWave, T, 0, stream>>>(bufG, norm, src, dst, bufA, E);

  // 3) layer 2: g2 = relu(a1) @ W2 ; a2 = segsum(...) + selfloop + b2
  k_gemm_wmma<1><<<gGemm, 128, 0, stream>>>(bufA, W2, nullptr, bufG, N);
  k_agg_init<<<gNF, T, 0, stream>>>(bufG, dinv, b2, bufA, N * FDIM);
  k_scatter<<<gEdgeWave, T, 0, stream>>>(bufG, norm, src, dst, bufA, E);

  // 4) out = relu(a2) @ Wfc + bfc
  k_gemm_wmma<1><<<gGemm, 128, 0, stream>>>(bufA, Wfc, bfc, outF, N);
}